// VirtualAdaptiveWeights_43688407334935
// MI455X (gfx1250) — compile-verified
//
#include <hip/hip_runtime.h>

typedef __attribute__((ext_vector_type(2))) float v2f;
typedef __attribute__((ext_vector_type(8))) float v8f;

#define N_NODES  100000
#define N_EDGES  500000
#define DIM      128
#define E_DIM    64
#define TOTAL    320     // 2*DIM + E_DIM
#define HEADS    8
#define NPAD     16      // pad heads to WMMA N=16
#define NTILES   (N_EDGES / 16)   // 31250 exactly

// ---------------------------------------------------------------------------
// Kernel 1: fold W1@W2 -> Wc[320][16] (cols 8..15 zero) and b1@W2+b2 -> bc[16]
// Tiny (2560 length-320 dots); launch-overhead-level cost.
// ---------------------------------------------------------------------------
__global__ void fold_weights_kernel(const float* __restrict__ W1,
                                    const float* __restrict__ b1,
                                    const float* __restrict__ W2,
                                    const float* __restrict__ b2,
                                    float* __restrict__ Wc,
                                    float* __restrict__ bc) {
  int tid = blockIdx.x * blockDim.x + threadIdx.x;
  for (int idx = tid; idx < TOTAL * NPAD; idx += blockDim.x * gridDim.x) {
    int k = idx >> 4, n = idx & 15;
    float s = 0.f;
    if (n < HEADS) {
      const float* w1row = W1 + k * TOTAL;
      #pragma unroll 4
      for (int j = 0; j < TOTAL; ++j) s = fmaf(w1row[j], W2[j * HEADS + n], s);
    }
    Wc[idx] = s;
  }
  if (tid < NPAD) {
    float s = 0.f;
    if (tid < HEADS) {
      s = b2[tid];
      for (int j = 0; j < TOTAL; ++j) s = fmaf(b1[j], W2[j * HEADS + tid], s);
    }
    bc[tid] = s;
  }
}

// ---------------------------------------------------------------------------
// Kernel 2: per 16-edge tile, D(16x16,f32) += A(16x4) x B(4x16) over K=320
// using V_WMMA_F32_16X16X4_F32 (fp32-exact path). B comes from LDS-resident
// folded weights; A comes from the L2-resident node-feature gathers.
// ---------------------------------------------------------------------------
__global__ __launch_bounds__(256) void edge_gemm_kernel(
    const float* __restrict__ x,           // [N_NODES][DIM]
    const int*   __restrict__ edge_index,  // [2][N_EDGES]
    const float* __restrict__ emb,         // [N_EDGES][E_DIM]
    const float* __restrict__ Wc,          // [TOTAL][NPAD]
    const float* __restrict__ bc,          // [NPAD]
    float* __restrict__ out) {             // [N_EDGES][HEADS]
  __shared__ float sWc[TOTAL * NPAD];      // 20 KB
  __shared__ float sBc[NPAD];
  for (int i = threadIdx.x; i < TOTAL * NPAD; i += blockDim.x) sWc[i] = Wc[i];
  if (threadIdx.x < NPAD) sBc[threadIdx.x] = bc[threadIdx.x];
  __syncthreads();

  const int lane = threadIdx.x & 31;
  const int m    = lane & 15;   // A row (edge-in-tile) == B column (head) lane slot
  const int hi   = lane >> 4;   // K half-pair selector (K+0/1 vs K+2/3)
  const int koff = hi * 2;
  const int wavesPerBlock = blockDim.x >> 5;
  const int wid    = blockIdx.x * wavesPerBlock + (threadIdx.x >> 5);
  const int nWaves = gridDim.x * wavesPerBlock;

  for (int tile = wid; tile < NTILES; tile += nWaves) {
    const int e    = tile * 16 + m;
    const int orig = edge_index[e];
    const int dest = edge_index[N_EDGES + e];
    const float* po = x   + (size_t)orig * DIM;
    const float* pd = x   + (size_t)dest * DIM;
    const float* pe = emb + (size_t)e * E_DIM;

    v8f acc = {0.f, 0.f, 0.f, 0.f, 0.f, 0.f, 0.f, 0.f};

    #pragma unroll 8
    for (int it = 0; it < 32; ++it) {            // K in [0,128): x[origin]
      const int kg = it * 4 + koff;
      v2f a = *(const v2f*)(po + kg);
      v2f b; b.x = sWc[kg * NPAD + m]; b.y = sWc[(kg + 1) * NPAD + m];
      acc = __builtin_amdgcn_wmma_f32_16x16x4_f32(false, a, false, b,
                                                  (short)0, acc, false, false);
    }
    #pragma unroll 8
    for (int it = 0; it < 32; ++it) {            // K in [128,256): x[dest]
      const int kg = 128 + it * 4 + koff;
      v2f a = *(const v2f*)(pd + (kg - 128));
      v2f b; b.x = sWc[kg * NPAD + m]; b.y = sWc[(kg + 1) * NPAD + m];
      acc = __builtin_amdgcn_wmma_f32_16x16x4_f32(false, a, false, b,
                                                  (short)0, acc, false, false);
    }
    #pragma unroll 8
    for (int it = 0; it < 16; ++it) {            // K in [256,320): edge emb
      const int kg = 256 + it * 4 + koff;
      v2f a = *(const v2f*)(pe + (kg - 256));
      v2f b; b.x = sWc[kg * NPAD + m]; b.y = sWc[(kg + 1) * NPAD + m];
      acc = __builtin_amdgcn_wmma_f32_16x16x4_f32(false, a, false, b,
                                                  (short)0, acc, false, false);
    }

    // D layout: lane m = N (head), VGPR r -> M = r + 8*hi
    if (m < HEADS) {
      const float bias = sBc[m];
      #pragma unroll
      for (int r = 0; r < 8; ++r) {
        const int row = tile * 16 + r + hi * 8;
        out[row * HEADS + m] = acc[r] + bias;
      }
    }
  }
}

extern "C" void kernel_launch(void* const* d_in, const int* in_sizes, int n_in,
                              void* d_out, int out_size, void* d_ws, size_t ws_size,
                              hipStream_t stream) {
  const float* x   = (const float*)d_in[0];
  const int*   ei  = (const int*)d_in[1];
  const float* emb = (const float*)d_in[2];
  const float* W1  = (const float*)d_in[3];
  const float* b1  = (const float*)d_in[4];
  const float* W2  = (const float*)d_in[5];
  const float* b2  = (const float*)d_in[6];
  float* out = (float*)d_out;

  float* Wc = (float*)d_ws;                 // 320*16 floats = 20480 B
  float* bc = Wc + TOTAL * NPAD;            // 16 floats

  fold_weights_kernel<<<40, 128, 0, stream>>>(W1, b1, W2, b2, Wc, bc);
  edge_gemm_kernel<<<1024, 256, 0, stream>>>(x, ei, emb, Wc, bc, out);
}